// RenderPipeline_50105088475645
// MI455X (gfx1250) — compile-verified
//
#include <hip/hip_runtime.h>

// MI455X (gfx1250) streaming affine point-cloud kernel.
// Memory-bound (~266 MB @ 23.3 TB/s => ~11.4 us floor). Data-movement path:
//   depth  --TDM tensor_load_to_lds-->  LDS  --ds_load_b128--> VGPRs
//   output --non-temporal global_store_b128 (201 MB > 192 MB L2, bypass)

typedef __attribute__((ext_vector_type(4))) float        v4f;
typedef __attribute__((ext_vector_type(4))) unsigned int v4u;
typedef __attribute__((ext_vector_type(4))) int          v4i;
typedef __attribute__((ext_vector_type(8))) int          v8i;

#if __has_include(<hip/amd_detail/amd_gfx1250_TDM.h>)
#define ATHENA_TDM_6ARG 1   // therock / clang-23 headers: 6-arg builtin
#endif

#define THREADS 256
#define PX_PER_BLOCK 1024   // 4 px / thread

__global__ __launch_bounds__(THREADS) void render_org_pc_kernel(
    const float* __restrict__ depth,    // (B,1,64,64)
    const float* __restrict__ rotates,  // (B,3)
    const float* __restrict__ trans,    // (B,3)
    float* __restrict__ out)            // (B,3,64,64)
{
    __shared__ float s_depth[PX_PER_BLOCK];
    __shared__ float s_cst[12];         // R[9], t'[3]

    const int blk = blockIdx.x;
    const int b   = blk >> 2;           // batch index
    const int pb  = (blk & 3) << 10;    // pixel base within image (0/1024/2048/3072)
    const int t   = threadIdx.x;

    if (t == 0) {
        // ---------- TDM: DMA this block's 4 KB depth slice into LDS ----------
        unsigned long long ga =
            (unsigned long long)(const void*)(depth + ((unsigned long long)b << 12)
                                              + (unsigned)pb);
        unsigned int la = (unsigned int)(unsigned long long)(&s_depth[0]); // LDS offset

        // D# group 0: count=1 valid, lds_addr, 57-bit global addr, type=2
        v4u g0 = { 1u,
                   la,
                   (unsigned int)(ga & 0xffffffffull),
                   (unsigned int)((ga >> 32) & 0x01ffffffull) | 0x80000000u };
        // D# group 1: data_size=4B, tensor_dim0=1024, tensor_dim1=1,
        //             tile_dim0=1024, tile_dim1=1, tensor_dim0_stride=1024
        v8i g1 = { (int)0x00020000,
                   (int)(1024u << 16),
                   (int)(1u << 16),
                   1024,
                   1,
                   1024,
                   0, 0 };
        v4i gz = {0, 0, 0, 0};
#ifdef ATHENA_TDM_6ARG
        v8i gz8 = {0, 0, 0, 0, 0, 0, 0, 0};
        __builtin_amdgcn_tensor_load_to_lds(g0, g1, gz, gz, gz8, 0);
#else
        __builtin_amdgcn_tensor_load_to_lds(g0, g1, gz, gz, 0);
#endif

        // ---------- per-batch R and t' (keeps reference's -sin(alpha) bug) ----------
        const float d2r = 0.017453292519943295f;
        float al = rotates[b * 3 + 2] * d2r;   // alpha = rot[:,2]
        float be = rotates[b * 3 + 0] * d2r;   // beta  = rot[:,0]
        float gm = rotates[b * 3 + 1] * d2r;   // gamma = rot[:,1]
        float sa, ca, sb, cb, sg, cg;
        sincosf(al, &sa, &ca);
        sincosf(be, &sb, &cb);
        sincosf(gm, &sg, &cg);
        // R = Rz(a) * Rx(b) * [cg 0 sg; 0 1 0; -sa 0 cg]  (bug: -sa, not -sg)
        float R00 = ca * cg - sa * sa * sb;
        float R01 = -sa * cb;
        float R02 = ca * sg + sa * sb * cg;
        float R10 = sa * cg + ca * sa * sb;
        float R11 = ca * cb;
        float R12 = sa * sg - ca * sb * cg;
        float R20 = -sa * cb;
        float R21 = sb;
        float R22 = cb * cg;
        s_cst[0] = R00; s_cst[1] = R01; s_cst[2] = R02;
        s_cst[3] = R10; s_cst[4] = R11; s_cst[5] = R12;
        s_cst[6] = R20; s_cst[7] = R21; s_cst[8] = R22;
        // t' = center + trans - R*center, center = (0,0,1)
        s_cst[9]  = trans[b * 3 + 0] - R02;
        s_cst[10] = trans[b * 3 + 1] - R12;
        s_cst[11] = 1.0f + trans[b * 3 + 2] - R22;

        __builtin_amdgcn_s_wait_tensorcnt(0);  // depth tile resident in LDS
    }
    __syncthreads();

    // 4 consecutive pixels per thread: p = x*64 + y, same x, y..y+3
    const int p0 = pb + (t << 2);
    const int x  = p0 >> 6;
    const int y0 = p0 & 63;
    const float invf = 2.7774178e-3f;          // 1/f, f = 63/(2*tan(5 deg))
    const float rx   = ((float)x  - 31.5f) * invf;
    const float ry0  = ((float)y0 - 31.5f) * invf;

    const float R00 = s_cst[0], R01 = s_cst[1], R02 = s_cst[2];
    const float R10 = s_cst[3], R11 = s_cst[4], R12 = s_cst[5];
    const float R20 = s_cst[6], R21 = s_cst[7], R22 = s_cst[8];
    const float tx  = s_cst[9], ty  = s_cst[10], tz = s_cst[11];

    // hoist the x-dependent part: q_n = R[n,0]*rx + R[n,2]
    const float q0 = fmaf(R00, rx, R02);
    const float q1 = fmaf(R10, rx, R12);
    const float q2 = fmaf(R20, rx, R22);

    const v4f d4 = *(const v4f*)&s_depth[t << 2];  // ds_load_b128

    v4f ox, oy, oz;
#pragma unroll
    for (int j = 0; j < 4; ++j) {
        float ry = fmaf((float)j, invf, ry0);
        float dj = d4[j];
        ox[j] = fmaf(dj, fmaf(R01, ry, q0), tx);
        oy[j] = fmaf(dj, fmaf(R11, ry, q1), ty);
        oz[j] = fmaf(dj, fmaf(R21, ry, q2), tz);
    }

    // (B,3,64,64): planes 4096 floats apart; NT B128 stores, coalesced per wave
    float* o = out + (unsigned long long)b * 12288ull + (unsigned)p0;
    __builtin_nontemporal_store(ox, (v4f*)(o));
    __builtin_nontemporal_store(oy, (v4f*)(o + 4096));
    __builtin_nontemporal_store(oz, (v4f*)(o + 8192));
}

extern "C" void kernel_launch(void* const* d_in, const int* in_sizes, int n_in,
                              void* d_out, int out_size, void* d_ws, size_t ws_size,
                              hipStream_t stream) {
    (void)n_in; (void)out_size; (void)d_ws; (void)ws_size;
    const float* depth   = (const float*)d_in[0];
    const float* rotates = (const float*)d_in[1];
    const float* trans   = (const float*)d_in[2];
    float* out           = (float*)d_out;

    const int B = in_sizes[1] / 3;        // 4096
    dim3 grid(B * 4), block(THREADS);     // 1024 px / block
    render_org_pc_kernel<<<grid, block, 0, stream>>>(depth, rotates, trans, out);
}